// LBPropagator_43224550867363
// MI455X (gfx1250) — compile-verified
//
#include <hip/hip_runtime.h>
#include <cstdint>

// Flip to 0 if the async-to-LDS inline asm is rejected by the assembler.
#define USE_ASYNC_LDS 1

typedef __attribute__((ext_vector_type(2))) float v2f;
typedef __attribute__((ext_vector_type(8))) float v8f;

// -------------------------------------------------------------------------
// CDNA5 async global->LDS copy (VGLOBAL GLOBAL_LOAD_ASYNC_TO_LDS_B128).
// Per-lane: copies 16 bytes from the lane's global address to the lane's LDS
// byte offset. Tracked by ASYNCcnt. Generic LDS pointers carry the LDS byte
// offset in their low 32 bits (ISA 10.2 aperture layout), so truncation gives
// the VDST operand.
// -------------------------------------------------------------------------
__device__ __forceinline__ void async_copy_b128(const float* gsrc, const float* ldst)
{
    const uint32_t loff = (uint32_t)(uintptr_t)ldst;
    asm volatile("global_load_async_to_lds_b128 %0, %1, off"
                 :: "v"(loff), "v"(gsrc) : "memory");
}

template <int N>
__device__ __forceinline__ void wait_asynccnt()
{
    asm volatile("s_wait_asynccnt %0" :: "i"(N) : "memory");
}

// XOR-swizzled LDS layouts: conflict-free fragment reads, 16B-contiguous
// destinations for async b128 beats (no padding).
__device__ __forceinline__ int aidx(int m, int k)   // A tile: BM x 16
{
    return m * 16 + ((((k >> 2) ^ ((m >> 2) & 3)) << 2) | (k & 3));
}
__device__ __forceinline__ int bidx(int k, int n)   // B tile: 16 x 64
{
    return k * 64 + ((((n >> 2) ^ ((k & 2) << 1)) << 2) | (n & 3));
}

// -------------------------------------------------------------------------
// bounds_eval: y_lb[i] = sum_k max(L,0)x_lb + min(L,0)x_ub + l[i]
//              y_ub[i] = sum_k max(U,0)x_ub + min(U,0)x_lb + u[i]
// -------------------------------------------------------------------------
__global__ __launch_bounds__(256)
void bounds_eval_kernel(const float* __restrict__ Lm, const float* __restrict__ Um,
                        const float* __restrict__ lv, const float* __restrict__ uv,
                        const float* __restrict__ xlb, const float* __restrict__ xub,
                        int M, int K,
                        float* __restrict__ ylb, float* __restrict__ yub)
{
    const int wid  = threadIdx.x >> 5;
    const int lane = threadIdx.x & 31;
    const int row  = blockIdx.x * 8 + wid;
    if (row >= M) return;
    const float* Lr = Lm + (size_t)row * K;
    const float* Ur = Um + (size_t)row * K;
    float pl = 0.f, pu = 0.f;
    for (int k = lane; k < K; k += 32) {
        const float Lk = Lr[k], Uk = Ur[k];
        const float lb = xlb[k], ub = xub[k];
        pl += fmaxf(Lk, 0.f) * lb + fminf(Lk, 0.f) * ub;
        pu += fmaxf(Uk, 0.f) * ub + fminf(Uk, 0.f) * lb;
    }
    #pragma unroll
    for (int off = 16; off; off >>= 1) {
        pl += __shfl_xor(pl, off);
        pu += __shfl_xor(pu, off);
    }
    if (lane == 0) {
        ylb[row] = pl + lv[row];
        yub[row] = pu + uv[row];
    }
}

// -------------------------------------------------------------------------
// ReLU relaxation coefficients (matches reference incl. 1e-12 guard)
// -------------------------------------------------------------------------
__global__ void slopes_kernel(const float* __restrict__ ylb, const float* __restrict__ yub,
                              int n, float* __restrict__ ls, float* __restrict__ us,
                              float* __restrict__ uo)
{
    const int i = blockIdx.x * 256 + threadIdx.x;
    if (i >= n) return;
    const float lb = ylb[i], ub = yub[i];
    const bool active = (lb >= 0.f);
    const bool dead   = (ub <= 0.f);
    const float denom = ub - lb;
    const float safe  = (fabsf(denom) < 1e-12f) ? 1.f : denom;
    const float su    = ub / safe;
    const float up_s  = active ? 1.f : (dead ? 0.f : su);
    const float up_o  = (active || dead) ? 0.f : (-su * lb);
    const float lo_s  = active ? 1.f : (dead ? 0.f : ((fabsf(lb) < fabsf(ub)) ? su : 0.f));
    ls[i] = lo_s; us[i] = up_s; uo[i] = up_o;
}

// -------------------------------------------------------------------------
// dual matvec: outL[i] = sum_k A[i,k]*(sL[k]*aL[k])       + bias[i]
//              outU[i] = sum_k A[i,k]*(sU[k]*aU[k]+oU[k]) + bias[i]
// -------------------------------------------------------------------------
__global__ __launch_bounds__(256)
void matvec_dual_kernel(const float* __restrict__ A,
                        const float* __restrict__ aL, const float* __restrict__ sL,
                        const float* __restrict__ aU, const float* __restrict__ sU,
                        const float* __restrict__ oU, const float* __restrict__ bias,
                        int M, int K,
                        float* __restrict__ outL, float* __restrict__ outU)
{
    const int wid  = threadIdx.x >> 5;
    const int lane = threadIdx.x & 31;
    const int row  = blockIdx.x * 8 + wid;
    if (row >= M) return;
    const float* Ar = A + (size_t)row * K;
    float pl = 0.f, pu = 0.f;
    for (int k = lane; k < K; k += 32) {
        const float a = Ar[k];
        pl += a * (sL[k] * aL[k]);
        pu += a * (sU[k] * aU[k] + oU[k]);
    }
    #pragma unroll
    for (int off = 16; off; off >>= 1) {
        pl += __shfl_xor(pl, off);
        pu += __shfl_xor(pu, off);
    }
    if (lane == 0) {
        outL[row] = pl + bias[row];
        outU[row] = pu + bias[row];
    }
}

// -------------------------------------------------------------------------
// Dual GEMM, V_WMMA_F32_16X16X4_F32, async double-buffered LDS pipeline:
//   CL = A @ (diag(sL) * BL),  CU = A @ (diag(sU) * BU)
// 256 threads = 8 waves (WM_ x WN_). Block tile BM_ x 64, BK = 16.
// Tile t+1 is fetched with GLOBAL_LOAD_ASYNC_TO_LDS_B128 while tile t's
// WMMAs run; a counted s_wait_asynccnt separates the two tiles in flight.
// K-scales are applied at fragment-load time (broadcast from LDS).
// -------------------------------------------------------------------------
template <int BM_, int WM_, int WN_>
__global__ __launch_bounds__(256)
void gemm_dual_wmma(const float* __restrict__ A,
                    const float* __restrict__ BLm, const float* __restrict__ BUm,
                    const float* __restrict__ sL, const float* __restrict__ sU,
                    float* __restrict__ CL, float* __restrict__ CU,
                    int M, int N, int K)
{
    constexpr int MT  = BM_ / WM_ / 16;   // 16x16 C tiles per wave (M)
    constexpr int NT  = 64 / WN_ / 16;    // 16x16 C tiles per wave (N)
    constexpr int APT = BM_ / 64;         // A b128 beats per thread per tile
    constexpr int NASYNC = APT + 2;       // async instrs per thread per tile

    __shared__ float As [2][BM_ * 16];
    __shared__ float BsL[2][16 * 64];
    __shared__ float BsU[2][16 * 64];
    __shared__ float Ss [2][2][16];       // [buf][L/U][k] scales

    const int t    = threadIdx.x;
    const int lane = t & 31;
    const int wid  = t >> 5;
    const int wm   = wid / WN_;
    const int wn   = wid % WN_;
    const int m_blk = blockIdx.x * BM_;
    const int n_blk = blockIdx.y * 64;

    const int fr    = lane & 15;          // M row (A) / N col (B,C)
    const int kHi   = (lane >> 4) * 2;    // lanes 16-31 hold K=2,3
    const int cHalf = (lane >> 4) * 8;    // C: lanes 16-31 hold M=r+8
    const int wmOff = wm * (MT * 16);
    const int wnOff = wn * (NT * 16);

    const v8f vzero = {0.f,0.f,0.f,0.f,0.f,0.f,0.f,0.f};
    v8f accL[MT][NT], accU[MT][NT];
    #pragma unroll
    for (int i = 0; i < MT; ++i)
        #pragma unroll
        for (int j = 0; j < NT; ++j) { accL[i][j] = vzero; accU[i][j] = vzero; }

    auto stage = [&](int kt, int buf) {
        #pragma unroll
        for (int i = 0; i < APT; ++i) {                 // A tile: BM_ x 16
            const int idx = t + i * 256;
            const int m = idx >> 2, g = idx & 3;
            const float* src = A + (size_t)(m_blk + m) * K + kt + g * 4;
            float* dst = &As[buf][aidx(m, g * 4)];
#if USE_ASYNC_LDS
            async_copy_b128(src, dst);
#else
            *(float4*)dst = *(const float4*)src;
#endif
        }
        {                                               // B tiles: 16 x 64
            const int k = t >> 4, ng = t & 15;
            const size_t srow = (size_t)(kt + k) * N + n_blk + ng * 4;
            const int d = bidx(k, ng * 4);
#if USE_ASYNC_LDS
            async_copy_b128(BLm + srow, &BsL[buf][d]);
            async_copy_b128(BUm + srow, &BsU[buf][d]);
#else
            *(float4*)&BsL[buf][d] = *(const float4*)(BLm + srow);
            *(float4*)&BsU[buf][d] = *(const float4*)(BUm + srow);
#endif
        }
        if (t < 32) {                                   // per-tile K scales
            const float* s = (t < 16) ? sL : sU;
            Ss[buf][t >> 4][t & 15] = s[kt + (t & 15)];
        }
    };

    const int T = K / 16;
    stage(0, 0);
    for (int ti = 0; ti < T; ++ti) {
        const int buf = ti & 1;
        const bool more = (ti + 1 < T);
        if (more) stage((ti + 1) * 16, buf ^ 1);
#if USE_ASYNC_LDS
        if (more) wait_asynccnt<NASYNC>();   // tile ti complete, ti+1 in flight
        else      wait_asynccnt<0>();
#endif
        __syncthreads();

        #pragma unroll
        for (int k0 = 0; k0 < 16; k0 += 4) {
            v2f afr[MT];
            #pragma unroll
            for (int mt = 0; mt < MT; ++mt) {
                const int m = wmOff + mt * 16 + fr;
                afr[mt] = *(const v2f*)&As[buf][aidx(m, k0 + kHi)];
            }
            const float slx = Ss[buf][0][k0 + kHi];
            const float sly = Ss[buf][0][k0 + kHi + 1];
            const float sux = Ss[buf][1][k0 + kHi];
            const float suy = Ss[buf][1][k0 + kHi + 1];
            v2f blfr[NT], bufr[NT];
            #pragma unroll
            for (int nt = 0; nt < NT; ++nt) {
                const int n  = wnOff + nt * 16 + fr;
                const int r0 = k0 + kHi;
                blfr[nt].x = BsL[buf][bidx(r0,     n)] * slx;
                blfr[nt].y = BsL[buf][bidx(r0 + 1, n)] * sly;
                bufr[nt].x = BsU[buf][bidx(r0,     n)] * sux;
                bufr[nt].y = BsU[buf][bidx(r0 + 1, n)] * suy;
            }
            #pragma unroll
            for (int mt = 0; mt < MT; ++mt)
                #pragma unroll
                for (int nt = 0; nt < NT; ++nt) {
                    accL[mt][nt] = __builtin_amdgcn_wmma_f32_16x16x4_f32(
                        false, afr[mt], false, blfr[nt], (short)0, accL[mt][nt], false, false);
                    accU[mt][nt] = __builtin_amdgcn_wmma_f32_16x16x4_f32(
                        false, afr[mt], false, bufr[nt], (short)0, accU[mt][nt], false, false);
                }
        }
        __syncthreads();
    }

    // ---- epilogue: C layout = 8 VGPRs, lanes 0-15 M=r, lanes 16-31 M=r+8 ----
    #pragma unroll
    for (int mt = 0; mt < MT; ++mt)
        #pragma unroll
        for (int nt = 0; nt < NT; ++nt) {
            const int gn = n_blk + wnOff + nt * 16 + fr;
            #pragma unroll
            for (int r = 0; r < 8; ++r) {
                const int gm = m_blk + wmOff + mt * 16 + cHalf + r;
                CL[(size_t)gm * N + gn] = accL[mt][nt][r];
                CU[(size_t)gm * N + gn] = accU[mt][nt][r];
            }
        }
}

// -------------------------------------------------------------------------
extern "C" void kernel_launch(void* const* d_in, const int* in_sizes, int n_in,
                              void* d_out, int out_size, void* d_ws, size_t ws_size,
                              hipStream_t stream)
{
    (void)n_in; (void)out_size; (void)ws_size;
    const float* x_lb = (const float*)d_in[0];
    const float* x_ub = (const float*)d_in[1];
    const float* W1   = (const float*)d_in[2];
    const float* b1   = (const float*)d_in[3];
    const float* W2   = (const float*)d_in[4];
    const float* b2   = (const float*)d_in[5];
    const float* W3   = (const float*)d_in[6];
    const float* b3   = (const float*)d_in[7];

    const int in_dim  = in_sizes[0];   // 1024
    const int hidden  = in_sizes[3];   // 4096
    const int out_dim = in_sizes[7];   // 1024

    float* ws = (float*)d_ws;
    size_t o = 0;
    float* L2  = ws + o; o += (size_t)hidden * in_dim;
    float* U2  = ws + o; o += (size_t)hidden * in_dim;
    float* L3  = ws + o; o += (size_t)out_dim * in_dim;
    float* U3  = ws + o; o += (size_t)out_dim * in_dim;
    float* ls1 = ws + o; o += hidden;
    float* us1 = ws + o; o += hidden;
    float* uo1 = ws + o; o += hidden;
    float* l2  = ws + o; o += hidden;
    float* u2  = ws + o; o += hidden;
    float* ls2 = ws + o; o += hidden;
    float* us2 = ws + o; o += hidden;
    float* uo2 = ws + o; o += hidden;
    float* l3  = ws + o; o += out_dim;
    float* u3  = ws + o; o += out_dim;
    float* y1l = ws + o; o += hidden;
    float* y1u = ws + o; o += hidden;
    float* y2l = ws + o; o += hidden;
    float* y2u = ws + o; o += hidden;

    float* y_lb = (float*)d_out;
    float* y_ub = y_lb + out_dim;

    // After layer 1: L = U = W1, l = u = b1 (W @ I = W algebraically).
    bounds_eval_kernel<<<hidden/8, 256, 0, stream>>>(W1, W1, b1, b1, x_lb, x_ub,
                                                     hidden, in_dim, y1l, y1u);
    slopes_kernel<<<(hidden+255)/256, 256, 0, stream>>>(y1l, y1u, hidden, ls1, us1, uo1);

    matvec_dual_kernel<<<hidden/8, 256, 0, stream>>>(W2, b1, ls1, b1, us1, uo1, b2,
                                                     hidden, hidden, l2, u2);

    // L2 = W2 @ diag(ls1) W1 ; U2 = W2 @ diag(us1) W1   (128x64 tiles, 512 WGs)
    gemm_dual_wmma<128, 4, 2><<<dim3(hidden/128, in_dim/64), 256, 0, stream>>>(
        W2, W1, W1, ls1, us1, L2, U2, hidden, in_dim, hidden);

    bounds_eval_kernel<<<hidden/8, 256, 0, stream>>>(L2, U2, l2, u2, x_lb, x_ub,
                                                     hidden, in_dim, y2l, y2u);
    slopes_kernel<<<(hidden+255)/256, 256, 0, stream>>>(y2l, y2u, hidden, ls2, us2, uo2);

    matvec_dual_kernel<<<out_dim/8, 256, 0, stream>>>(W3, l2, ls2, u2, us2, uo2, b3,
                                                      out_dim, hidden, l3, u3);

    // L3 = W3 @ diag(ls2) L2 ; U3 = W3 @ diag(us2) U2   (64x64 tiles, 256 WGs)
    gemm_dual_wmma<64, 2, 4><<<dim3(out_dim/64, in_dim/64), 256, 0, stream>>>(
        W3, L2, U2, ls2, us2, L3, U3, out_dim, in_dim, hidden);

    bounds_eval_kernel<<<out_dim/8, 256, 0, stream>>>(L3, U3, l3, u3, x_lb, x_ub,
                                                      out_dim, in_dim, y_lb, y_ub);
}